// QuadraticSplineRenderer_39092792329131
// MI455X (gfx1250) — compile-verified
//
#include <hip/hip_runtime.h>
#include <stdint.h>

#define IMG_HW   512
#define N_LINES  64
#define NPIX     (IMG_HW * IMG_HW)

typedef uint32_t u32x4 __attribute__((ext_vector_type(4)));
typedef uint32_t u32x8 __attribute__((ext_vector_type(8)));

// sign(x) * max(|x|,1e-12)^(1/3), matching jnp semantics (sign(0)=0)
__device__ __forceinline__ float cbrt_ref(float x) {
  float r = __expf(__logf(fmaxf(fabsf(x), 1e-12f)) * (1.0f/3.0f));
  r = __builtin_copysignf(r, x);
  return (x == 0.0f) ? 0.0f : r;
}

// ---------------------------------------------------------------------------
// Render kernel. Wave 0 of each block DMAs the 4 KB curve-constant table into
// LDS with the Tensor Data Mover (1-row D#: 1024 x f32), waits TENSORcnt,
// barrier; then all 8 wave32s consume it via broadcast ds_load_b128 at
// wave-uniform addresses while the trans pipes (sqrt/sin/cos/exp/log) work.
// The REAL LDS offset of sC is fed into the descriptor (g0[1]) — this both
// targets the table correctly and captures sC's address, so the optimizer
// must assume the asm writes it (no more folding the loop to poison).
// ---------------------------------------------------------------------------
__global__ void __launch_bounds__(256)
spline_render(const float* __restrict__ tbl, float* __restrict__ out) {
  __shared__ __align__(16) float sC[N_LINES * 16];

  if (threadIdx.x < 32u) {             // wave 0 issues the TDM op (scalar)
    uint64_t gaddr   = (uint64_t)(uintptr_t)tbl;
    uint32_t lds_off = (uint32_t)(uintptr_t)(void*)&sC[0];  // low 32 = LDS byte offset
    const uint32_t elems = N_LINES * 16;   // 1024 dwords
    u32x4 g0, g2, g3;
    u32x8 g1;
    // D# group 0: count=1, lds_addr=offset(sC), global_addr, type=2 ("image")
    g0[0] = 1u;
    g0[1] = lds_off;
    g0[2] = (uint32_t)gaddr;
    g0[3] = (uint32_t)((gaddr >> 32) & 0x01FFFFFFu) | (2u << 30);
    // D# group 1: workgroup_mask=0 (not in cluster), data_size=4B (code 2),
    // tensor_dim0 = tile_dim0 = 1024, tensor_dim1 = tile_dim1 = 1
    g1[0] = (2u << 16);
    g1[1] = (elems & 0xFFFFu) << 16;           // tensor_dim0[15:0]
    g1[2] = (elems >> 16) | (1u << 16);        // tensor_dim0[31:16] | tensor_dim1 lo
    g1[3] = (elems & 0xFFFFu) << 16;           // tensor_dim1 hi=0 | tile_dim0
    g1[4] = 1u;                                // tile_dim1=1, tile_dim2 unused
    g1[5] = elems;                             // tensor_dim0_stride lo32
    g1[6] = 0u;
    g1[7] = 0u;
    // Groups 2/3: higher dims degenerate
    g2[0] = 1u; g2[1] = 0u; g2[2] = 0u; g2[3] = 0u;
    g3[0] = 0u; g3[1] = 0u; g3[2] = 0u; g3[3] = 0u;
    asm volatile("tensor_load_to_lds %0, %1, %2, %3"
                 :: "s"(g0), "s"(g1), "s"(g2), "s"(g3)
                 : "memory");
    __builtin_amdgcn_s_wait_tensorcnt(0);
  }
  __syncthreads();

  // Opaque zero index: keeps the element selection unanalyzable (defense in
  // depth) without pulling the loads out of addrspace(3).
  unsigned lbase = 0;
  asm volatile("" : "+s"(lbase));

  int g = blockIdx.x * blockDim.x + threadIdx.x;
  int i = g >> 9;                 // index along first meshgrid axis (W)
  int j = g & (IMG_HW - 1);       // index along second axis (H)
  const float invN = 1.0f / (float)(IMG_HW - 1);
  float px = (float)i * invN;     // p = (yy, xx) per reference
  float py = (float)j * invN;

  float acc = 0.0f;
  #pragma unroll 2
  for (int k = 0; k < N_LINES; ++k) {
    const float4* qv = reinterpret_cast<const float4*>(&sC[lbase + k * 16]);
    float4 q0 = qv[0];            // Ax Ay Bx By
    float4 q1 = qv[1];            // ax ay ca cb0
    float4 q2 = qv[2];            // cc0 Bix Biy Aix
    float4 q3 = qv[3];            // Aiy lw lc pad
    float Ax  = q0.x, Ay  = q0.y, Bx  = q0.z, By  = q0.w;
    float ax_ = q1.x, ay_ = q1.y, ca  = q1.z, cb0 = q1.w;
    float cc0 = q2.x, Bix = q2.y, Biy = q2.z, Aix = q2.w;
    float Aiy = q3.x, lwv = q3.y, lcv = q3.z;

    float cb = cb0 - (px*Bix + py*Biy);       // cubic ay
    float cc = cc0 - (px*Aix + py*Aiy);       // cubic az

    // solve_cubic(ca, cb, cc)
    float pco = cb - ca*ca*(1.0f/3.0f);
    float p3  = pco*pco*pco;
    float qc  = ca*(2.0f*ca*ca - 9.0f*cb)*(1.0f/27.0f) + cc;
    float d   = qc*qc + p3*(4.0f/27.0f);
    float sq  = __builtin_sqrtf(fmaxf(d, 1e-12f));
    float root0 = cbrt_ref(( sq - qc)*0.5f)
                + cbrt_ref((-sq - qc)*0.5f) - ca*(1.0f/3.0f);

    float p3s = (fabsf(p3) > 1e-9f) ? p3 : 1e-9f;
    float arg = -__builtin_sqrtf(fmaxf(-27.0f/p3s, 1e-12f)) * qc * 0.5f;
    arg = fminf(fmaxf(arg, -1.0f + 1e-7f), 1.0f - 1e-7f);
    float v  = acosf(arg) * (1.0f/3.0f);
    float m  = __cosf(v);
    float n  = __sinf(v) * 1.7320508075688772f;
    float sp = __builtin_sqrtf(fmaxf(-pco*(1.0f/3.0f), 1e-12f));
    float r1 = (m + m)*sp - ca*(1.0f/3.0f);
    float r2 = (-n - m)*sp - ca*(1.0f/3.0f);

    float t0 = (d > 0.0f) ? root0 : r1;
    float t1 = (d > 0.0f) ? root0 : r2;
    t0 = fminf(fmaxf(t0, 0.0f), 1.0f);
    t1 = fminf(fmaxf(t1, 0.0f), 1.0f);

    float Cx = px - ax_, Cy = py - ay_;
    float Dx = Ax + Ax,  Dy = Ay + Ay;
    float d1x = (Dx + Bx*t0)*t0 - Cx, d1y = (Dy + By*t0)*t0 - Cy;
    float d2x = (Dx + Bx*t1)*t1 - Cx, d2y = (Dy + By*t1)*t1 - Cy;
    float dd  = fminf(d1x*d1x + d1y*d1y, d2x*d2x + d2y*d2y);
    float dist = __builtin_sqrtf(fmaxf(dd, 1e-12f));

    float s = (lwv - dist) / lwv * 6.0f;
    acc += lcv / (1.0f + __expf(-s));         // sigmoid * lc
  }

  if (g < NPIX)
    out[g] = 1.0f - fminf(fmaxf(acc, 0.0f), 1.0f);
}

// ---------------------------------------------------------------------------
// Per-curve constants (16 floats per curve, 4 KB total) into d_ws.
//   cubic_a = 3(A.B)/BB                       (pixel-independent)
//   cubic_b = cb0 - p.(B/BB),  cb0 = (2(A.A) + a.B)/BB
//   cubic_c = cc0 - p.(A/BB),  cc0 = (a.A)/BB
// ---------------------------------------------------------------------------
__global__ void curve_precompute(const float* __restrict__ a,
                                 const float* __restrict__ b,
                                 const float* __restrict__ c,
                                 const float* __restrict__ lw,
                                 const float* __restrict__ lc,
                                 float* __restrict__ tbl) {
  int i = threadIdx.x;
  if (i >= N_LINES) return;
  float axp = a[2*i],          ayp = a[2*i+1];
  float bxp = b[2*i] + 1e-5f,  byp = b[2*i+1] + 1e-5f;
  float cxp = c[2*i],          cyp = c[2*i+1];
  float Ax = bxp - axp,        Ay = byp - ayp;
  float Bx = cxp - bxp - Ax,   By = cyp - byp - Ay;
  float BB  = Bx*Bx + By*By;
  float inv = 1.0f / BB;
  float AB  = Ax*Bx + Ay*By;
  float AA  = Ax*Ax + Ay*Ay;
  float aB  = axp*Bx + ayp*By;
  float aA  = axp*Ax + ayp*Ay;
  float* o = tbl + 16*i;
  o[0]  = Ax;               o[1]  = Ay;
  o[2]  = Bx;               o[3]  = By;
  o[4]  = axp;              o[5]  = ayp;
  o[6]  = 3.0f*AB*inv;      o[7]  = (2.0f*AA + aB)*inv;
  o[8]  = aA*inv;           o[9]  = Bx*inv;
  o[10] = By*inv;           o[11] = Ax*inv;
  o[12] = Ay*inv;           o[13] = lw[i];
  o[14] = lc[i];            o[15] = 0.0f;
}

// ---------------------------------------------------------------------------
extern "C" void kernel_launch(void* const* d_in, const int* in_sizes, int n_in,
                              void* d_out, int out_size, void* d_ws, size_t ws_size,
                              hipStream_t stream) {
  const float* a  = (const float*)d_in[0];
  const float* b  = (const float*)d_in[1];
  const float* c  = (const float*)d_in[2];
  const float* lw = (const float*)d_in[3];
  const float* lc = (const float*)d_in[4];
  float* tbl = (float*)d_ws;                 // 4 KB curve-constant table
  float* out = (float*)d_out;

  curve_precompute<<<1, 64, 0, stream>>>(a, b, c, lw, lc, tbl);
  spline_render<<<NPIX / 256, 256, 0, stream>>>(tbl, out);
}